// ClustGeoEdgeEncoder_15169824489856
// MI455X (gfx1250) — compile-verified
//
#include <hip/hip_runtime.h>

typedef float v2f __attribute__((ext_vector_type(2)));
typedef float v8f __attribute__((ext_vector_type(8)));

#define P_PTS 64
#define WAVES_PER_BLOCK 8

__global__ __launch_bounds__(256) void clust_geo_edge_kernel(
    const float* __restrict__ data,        // (N_POINTS, 4) f32
    const int*   __restrict__ clusts,      // (4096, 64) int
    const int*   __restrict__ edge_index,  // (2, E) int
    float*       __restrict__ out,         // (E, 19) f32
    int E)
{
    __shared__ float sx1[WAVES_PER_BLOCK][P_PTS * 3];
    __shared__ float sx2[WAVES_PER_BLOCK][P_PTS * 3];
    __shared__ float sn1[WAVES_PER_BLOCK][P_PTS];
    __shared__ float sn2[WAVES_PER_BLOCK][P_PTS];

    const int w    = threadIdx.x >> 5;
    const int lane = threadIdx.x & 31;
    int e = blockIdx.x * WAVES_PER_BLOCK + w;
    if (e >= E) e = E - 1;   // keep all waves alive (barrier-safe), redundant deterministic work

    const int c1 = edge_index[e];
    const int c2 = edge_index[E + e];

    // ---- Stage gathered coords + squared norms into LDS (2 points per lane) ----
    #pragma unroll
    for (int tt = 0; tt < 2; ++tt) {
        int t  = lane + tt * 32;
        int p1 = clusts[c1 * P_PTS + t];
        int p2 = clusts[c2 * P_PTS + t];
        float ax = data[p1 * 4 + 1], ay = data[p1 * 4 + 2], az = data[p1 * 4 + 3];
        float bx = data[p2 * 4 + 1], by = data[p2 * 4 + 2], bz = data[p2 * 4 + 3];
        sx1[w][t * 3 + 0] = ax; sx1[w][t * 3 + 1] = ay; sx1[w][t * 3 + 2] = az;
        sx2[w][t * 3 + 0] = bx; sx2[w][t * 3 + 1] = by; sx2[w][t * 3 + 2] = bz;
        sn1[w][t] = ax * ax + ay * ay + az * az;
        sn2[w][t] = bx * bx + by * by + bz * bz;
    }
    __syncthreads();

    const int half = lane >> 4;            // 0: holds K=0,1   1: holds K=2,3(pad)
    const int l16  = lane & 15;
    const float padm = half ? 0.0f : 1.0f; // zero the K-pad slot without predicating loads

    // ---- Preload B fragments (x2 columns, K-pad masked) and column norms ----
    v2f   bfrag[4];
    float n2v[4];
    #pragma unroll
    for (int tj = 0; tj < 4; ++tj) {
        int n = tj * 16 + l16;
        bfrag[tj].x = sx2[w][n * 3 + 2 * half];   // K=0 (half 0) / K=2 (half 1), address select
        bfrag[tj].y = padm * sx2[w][n * 3 + 1];   // K=1 / K=3 pad -> 0
        n2v[tj]     = sn2[w][n];
    }

    unsigned long long best = ~0ull;

    // ---- 4x4 tiles of 16x16: D = (-2*x1) x x2^T + (n1+n2) == d^2 straight from WMMA ----
    #pragma unroll
    for (int ti = 0; ti < 4; ++ti) {
        int m = ti * 16 + l16;
        v2f a;
        a.x = -2.0f * sx1[w][m * 3 + 2 * half];
        a.y = -2.0f * padm * sx1[w][m * 3 + 1];

        float n1r[8];
        #pragma unroll
        for (int r = 0; r < 8; ++r) n1r[r] = sn1[w][ti * 16 + half * 8 + r];

        #pragma unroll
        for (int tj = 0; tj < 4; ++tj) {
            v8f c;
            #pragma unroll
            for (int r = 0; r < 8; ++r) c[r] = n1r[r] + n2v[tj];

            // 8 args: (neg_a, A, neg_b, B, c_mod, C, reuse_a, reuse_b)
            v8f acc = __builtin_amdgcn_wmma_f32_16x16x4_f32(
                false, a, false, bfrag[tj], (short)0, c, false, false);

            #pragma unroll
            for (int r = 0; r < 8; ++r) {
                // Integer-domain clamp: max_i32(bits, 0) maps any negative float
                // (cancellation residue) to +0.0 bits in a single v_max_i32.
                int b = __float_as_int(acc[r]);
                b = (b < 0) ? 0 : b;
                int M = ti * 16 + half * 8 + r;     // D-tile row
                int N = tj * 16 + l16;              // D-tile col
                unsigned long long key =
                    ((unsigned long long)(unsigned)b << 32) |
                    (unsigned)(M * 64 + N);          // first-min tie rule == smallest flat idx
                best = (key < best) ? key : best;
            }
        }
    }

    // ---- wave32 butterfly min-reduction on 64-bit keys ----
    #pragma unroll
    for (int off = 16; off > 0; off >>= 1) {
        unsigned int lo = (unsigned int)best;
        unsigned int hi = (unsigned int)(best >> 32);
        lo = __shfl_xor(lo, off, 32);
        hi = __shfl_xor(hi, off, 32);
        unsigned long long o = ((unsigned long long)hi << 32) | lo;
        best = (o < best) ? o : best;
    }

    if (lane == 0) {
        int flat = (int)(best & 0xFFFFFFFFull);
        int i1 = flat >> 6;
        int i2 = flat & 63;
        float v1x = sx1[w][i1 * 3 + 0], v1y = sx1[w][i1 * 3 + 1], v1z = sx1[w][i1 * 3 + 2];
        float v2x = sx2[w][i2 * 3 + 0], v2y = sx2[w][i2 * 3 + 1], v2z = sx2[w][i2 * 3 + 2];
        float dx = v1x - v2x, dy = v1y - v2y, dz = v1z - v2z;
        float lend = sqrtf(dx * dx + dy * dy + dz * dz);
        float nx = dx, ny = dy, nz = dz;
        if (lend > 0.0f) { nx = dx / lend; ny = dy / lend; nz = dz / lend; }

        float* o = out + (size_t)e * 19;
        o[0]  = v1x; o[1]  = v1y; o[2]  = v1z;
        o[3]  = v2x; o[4]  = v2y; o[5]  = v2z;
        o[6]  = nx;  o[7]  = ny;  o[8]  = nz;
        o[9]  = lend;
        o[10] = nx * nx; o[11] = nx * ny; o[12] = nx * nz;
        o[13] = ny * nx; o[14] = ny * ny; o[15] = ny * nz;
        o[16] = nz * nx; o[17] = nz * ny; o[18] = nz * nz;
    }
}

extern "C" void kernel_launch(void* const* d_in, const int* in_sizes, int n_in,
                              void* d_out, int out_size, void* d_ws, size_t ws_size,
                              hipStream_t stream) {
    const float* data       = (const float*)d_in[0];
    const int*   clusts     = (const int*)d_in[1];
    const int*   edge_index = (const int*)d_in[2];
    float*       out        = (float*)d_out;

    const int E = in_sizes[2] / 2;   // edge_index is (2, E)
    const int blocks = (E + WAVES_PER_BLOCK - 1) / WAVES_PER_BLOCK;

    clust_geo_edge_kernel<<<dim3(blocks), dim3(256), 0, stream>>>(
        data, clusts, edge_index, out, E);
}